// GCNIINet_62423054680288
// MI455X (gfx1250) — compile-verified
//
#include <hip/hip_runtime.h>
#include <math.h>

#define NN      50000
#define NE      800000
#define IN_CH   512
#define HID     64
#define OUT_CH  40
#define NLAYERS 64

typedef __attribute__((ext_vector_type(2))) float v2f;
typedef __attribute__((ext_vector_type(8))) float v8f;

__device__ __forceinline__ v8f wmma4(v2f a, v2f b, v8f c) {
    // D = A(16x4,f32) * B(4x16,f32) + C(16x16,f32)
    return __builtin_amdgcn_wmma_f32_16x16x4_f32(false, a, false, b, (short)0, c,
                                                 false, false);
}

// ---------------- setup kernels ----------------

__global__ void k_zero(int* __restrict__ ecnt, int* __restrict__ cursor) {
    int i = blockIdx.x * blockDim.x + threadIdx.x;
    if (i < NN) { ecnt[i] = 0; cursor[i] = 0; }
}

__global__ void k_count(const long long* __restrict__ ei, int* __restrict__ ecnt) {
    int e = blockIdx.x * blockDim.x + threadIdx.x;
    if (e < NE) {
        int d = (int)ei[NE + e];
        atomicAdd(&ecnt[d], 1);
    }
}

__global__ void k_dinv(const int* __restrict__ ecnt, float* __restrict__ dinv) {
    int i = blockIdx.x * blockDim.x + threadIdx.x;
    if (i < NN) dinv[i] = rsqrtf((float)ecnt[i] + 1.0f);  // +1 self loop, deg>=1
}

// single-workgroup exclusive scan of ecnt -> offs[NN+1]
__global__ void k_scan(const int* __restrict__ ecnt, int* __restrict__ offs) {
    __shared__ int sdata[1024];
    __shared__ int carry;
    const int tid = threadIdx.x;
    if (tid == 0) carry = 0;
    __syncthreads();
    for (int base = 0; base < NN; base += 1024) {
        int i = base + tid;
        int v = (i < NN) ? ecnt[i] : 0;
        sdata[tid] = v;
        __syncthreads();
        for (int off = 1; off < 1024; off <<= 1) {
            int t = (tid >= off) ? sdata[tid - off] : 0;
            __syncthreads();
            sdata[tid] += t;
            __syncthreads();
        }
        if (i < NN) offs[i] = carry + sdata[tid] - v;  // exclusive
        __syncthreads();
        if (tid == 1023) carry += sdata[1023];
        __syncthreads();
    }
    if (tid == 0) offs[NN] = carry;
}

__global__ void k_fill(const long long* __restrict__ ei, const float* __restrict__ dinv,
                       const int* __restrict__ offs, int* __restrict__ cursor,
                       int* __restrict__ csrs, float* __restrict__ csrw) {
    int e = blockIdx.x * blockDim.x + threadIdx.x;
    if (e >= NE) return;
    int s = (int)ei[e];
    int d = (int)ei[NE + e];
    int p = offs[d] + atomicAdd(&cursor[d], 1);
    csrs[p] = s;
    csrw[p] = 0.5f * dinv[s] * dinv[d];  // (1-ALPHA) folded in
}

// canonical per-node order (by src) -> deterministic fp summation
__global__ void k_sortcsr(const int* __restrict__ offs, int* __restrict__ csrs,
                          float* __restrict__ csrw) {
    int n = blockIdx.x * blockDim.x + threadIdx.x;
    if (n >= NN) return;
    int b = offs[n], e = offs[n + 1];
    for (int i = b + 1; i < e; ++i) {
        int s = csrs[i]; float w = csrw[i];
        int j = i - 1;
        while (j >= b && csrs[j] > s) {
            csrs[j + 1] = csrs[j]; csrw[j + 1] = csrw[j]; --j;
        }
        csrs[j + 1] = s; csrw[j + 1] = w;
    }
}

// ---------------- lin1: h = x0 = relu(X @ W1 + b1) ----------------
// one wave per 16-node tile, wave computes 16x64 output slab via WMMA f32 16x16x4

__global__ __launch_bounds__(256) void k_lin1(const float* __restrict__ X,
                                              const float* __restrict__ W1,
                                              const float* __restrict__ b1,
                                              float* __restrict__ h,
                                              float* __restrict__ x0) {
    __shared__ float WsT[HID * 68];  // transposed, padded: WsT[n*68+k]
    const int lane = threadIdx.x & 31;
    const int wave = (blockIdx.x << 3) + (threadIdx.x >> 5);
    int tile = wave; if (tile > NN / 16 - 1) tile = NN / 16 - 1;  // clamp (benign dup)
    const int m0   = tile * 16;
    const int row  = lane & 15;
    const int kg   = (lane >> 4) << 1;  // 0 or 2
    const int ncol = lane & 15;

    v8f acc0 = {}, acc1 = {}, acc2 = {}, acc3 = {};

    for (int kc = 0; kc < IN_CH; kc += HID) {
        __syncthreads();
        for (int i = threadIdx.x; i < HID * HID; i += 256) {
            int kk = i >> 6, nn = i & 63;
            WsT[nn * 68 + kk] = W1[(kc + kk) * HID + nn];
        }
        __syncthreads();
        const float* arow = &X[(size_t)(m0 + row) * IN_CH + kc + kg];
#pragma unroll
        for (int k0 = 0; k0 < HID; k0 += 4) {
            v2f a  = *(const v2f*)&arow[k0];
            v2f b0 = *(const v2f*)&WsT[(0 + ncol) * 68 + k0 + kg];
            v2f bb1 = *(const v2f*)&WsT[(16 + ncol) * 68 + k0 + kg];
            v2f b2 = *(const v2f*)&WsT[(32 + ncol) * 68 + k0 + kg];
            v2f b3 = *(const v2f*)&WsT[(48 + ncol) * 68 + k0 + kg];
            acc0 = wmma4(a, b0, acc0);
            acc1 = wmma4(a, bb1, acc1);
            acc2 = wmma4(a, b2, acc2);
            acc3 = wmma4(a, b3, acc3);
        }
    }
    const int mh = (lane >> 4) * 8;
#pragma unroll
    for (int v = 0; v < 8; ++v) {
        const int m = m0 + mh + v;
        float* hr = &h[(size_t)m * HID];
        float* xr = &x0[(size_t)m * HID];
        float t;
        t = acc0[v] + b1[ncol];      t = t > 0.f ? t : 0.f; hr[ncol]      = t; xr[ncol]      = t;
        t = acc1[v] + b1[16 + ncol]; t = t > 0.f ? t : 0.f; hr[16 + ncol] = t; xr[16 + ncol] = t;
        t = acc2[v] + b1[32 + ncol]; t = t > 0.f ? t : 0.f; hr[32 + ncol] = t; xr[32 + ncol] = t;
        t = acc3[v] + b1[48 + ncol]; t = t > 0.f ? t : 0.f; hr[48 + ncol] = t; xr[48 + ncol] = t;
    }
}

// ---------------- per-layer aggregation ----------------
// z[i] = ALPHA*x0[i] + selfw*h[i] + sum_e csrw[e]*h[src[e]]
// one wave per node; lane owns 2 channels (float2) -> 256B coalesced row reads

__global__ __launch_bounds__(256) void k_aggz(const float* __restrict__ h,
                                              const float* __restrict__ x0,
                                              const float* __restrict__ dinv,
                                              const int* __restrict__ offs,
                                              const int* __restrict__ csrs,
                                              const float* __restrict__ csrw,
                                              float* __restrict__ z) {
    const int gid  = blockIdx.x * blockDim.x + threadIdx.x;
    const int node = gid >> 5;
    const int lane = threadIdx.x & 31;
    if (node >= NN) return;

    const float2* h2  = (const float2*)h;
    const float2* x02 = (const float2*)x0;
    float2*       z2  = (float2*)z;

    const float di    = dinv[node];
    const float selfw = 0.5f * di * di;  // (1-ALPHA)*dinv^2

    float2 xv = x02[(size_t)node * 32 + lane];
    float2 hv = h2[(size_t)node * 32 + lane];
    float ax = 0.5f * xv.x + selfw * hv.x;
    float ay = 0.5f * xv.y + selfw * hv.y;

    const int beg = offs[node], end = offs[node + 1];
    for (int e = beg; e < end; ++e) {
        int   s = csrs[e];
        float w = csrw[e];
        if (e + 1 < end)  // hide L2 latency of next random row
            __builtin_prefetch(&h2[(size_t)csrs[e + 1] * 32 + lane], 0, 1);
        float2 nv = h2[(size_t)s * 32 + lane];
        ax += w * nv.x;
        ay += w * nv.y;
    }
    float2 o; o.x = ax; o.y = ay;
    z2[(size_t)node * 32 + lane] = o;
}

// ---------------- per-layer GEMM: h = relu((1-beta)*z + beta*(z @ W)) ----------------

__global__ __launch_bounds__(256) void k_gemm_layer(const float* __restrict__ z,
                                                    const float* __restrict__ W,
                                                    float* __restrict__ h,
                                                    float beta) {
    __shared__ float WsT[HID * 68];
    const int lane = threadIdx.x & 31;
    const int wave = (blockIdx.x << 3) + (threadIdx.x >> 5);
    int tile = wave; if (tile > NN / 16 - 1) tile = NN / 16 - 1;
    const int m0   = tile * 16;
    const int row  = lane & 15;
    const int kg   = (lane >> 4) << 1;
    const int ncol = lane & 15;

    for (int i = threadIdx.x; i < HID * HID; i += 256) {
        int kk = i >> 6, nn = i & 63;
        WsT[nn * 68 + kk] = W[i];
    }
    __syncthreads();

    v8f acc0 = {}, acc1 = {}, acc2 = {}, acc3 = {};
    const float* arow = &z[(size_t)(m0 + row) * HID + kg];
#pragma unroll
    for (int k0 = 0; k0 < HID; k0 += 4) {
        v2f a  = *(const v2f*)&arow[k0];
        v2f b0 = *(const v2f*)&WsT[(0 + ncol) * 68 + k0 + kg];
        v2f b1 = *(const v2f*)&WsT[(16 + ncol) * 68 + k0 + kg];
        v2f b2 = *(const v2f*)&WsT[(32 + ncol) * 68 + k0 + kg];
        v2f b3 = *(const v2f*)&WsT[(48 + ncol) * 68 + k0 + kg];
        acc0 = wmma4(a, b0, acc0);
        acc1 = wmma4(a, b1, acc1);
        acc2 = wmma4(a, b2, acc2);
        acc3 = wmma4(a, b3, acc3);
    }

    const float omb = 1.0f - beta;
    const int   mh  = (lane >> 4) * 8;
#pragma unroll
    for (int v = 0; v < 8; ++v) {
        const int m = m0 + mh + v;
        const float* zr = &z[(size_t)m * HID];
        float*       hr = &h[(size_t)m * HID];
        float t;
        t = omb * zr[ncol]      + beta * acc0[v]; hr[ncol]      = t > 0.f ? t : 0.f;
        t = omb * zr[16 + ncol] + beta * acc1[v]; hr[16 + ncol] = t > 0.f ? t : 0.f;
        t = omb * zr[32 + ncol] + beta * acc2[v]; hr[32 + ncol] = t > 0.f ? t : 0.f;
        t = omb * zr[48 + ncol] + beta * acc3[v]; hr[48 + ncol] = t > 0.f ? t : 0.f;
    }
}

// ---------------- lin2 + log_softmax ----------------

__global__ __launch_bounds__(128) void k_out(const float* __restrict__ h,
                                             const float* __restrict__ W2,
                                             const float* __restrict__ b2,
                                             float* __restrict__ out) {
    __shared__ float hs[128 * 65];  // pad 65 to avoid bank conflicts
    const int n0 = blockIdx.x * 128;
    for (int i = threadIdx.x; i < 128 * 64; i += 128) {
        size_t g = (size_t)n0 * 64 + i;
        hs[(i >> 6) * 65 + (i & 63)] = (g < (size_t)NN * 64) ? h[g] : 0.f;
    }
    __syncthreads();

    const int n = n0 + threadIdx.x;
    if (n >= NN) return;
    const float* hr = &hs[threadIdx.x * 65];

    float o[OUT_CH];
#pragma unroll
    for (int j = 0; j < OUT_CH; ++j) {
        float acc = b2[j];
#pragma unroll 16
        for (int k = 0; k < HID; ++k) acc += hr[k] * W2[k * OUT_CH + j];
        o[j] = acc;
    }
    float m = o[0];
#pragma unroll
    for (int j = 1; j < OUT_CH; ++j) m = o[j] > m ? o[j] : m;
    float s = 0.f;
#pragma unroll
    for (int j = 0; j < OUT_CH; ++j) s += expf(o[j] - m);
    const float ls = logf(s) + m;
    float* orow = &out[(size_t)n * OUT_CH];
#pragma unroll
    for (int j = 0; j < OUT_CH; ++j) orow[j] = o[j] - ls;
}

// ---------------- host launcher ----------------

extern "C" void kernel_launch(void* const* d_in, const int* in_sizes, int n_in,
                              void* d_out, int out_size, void* d_ws, size_t ws_size,
                              hipStream_t stream) {
    const float*     x  = (const float*)d_in[0];
    const long long* ei = (const long long*)d_in[1];  // int64 [2, NE]
    const float*     w1 = (const float*)d_in[2];
    const float*     b1 = (const float*)d_in[3];
    const float*     cw = (const float*)d_in[4];      // [64, 64, 64]
    const float*     w2 = (const float*)d_in[5];
    const float*     b2 = (const float*)d_in[6];
    float* out = (float*)d_out;

    // workspace layout (all fp32/int32, base is 256B-aligned)
    float* h    = (float*)d_ws;                  // NN*64
    float* x0   = h    + (size_t)NN * HID;       // NN*64
    float* z    = x0   + (size_t)NN * HID;       // NN*64
    float* dinv = z    + (size_t)NN * HID;       // NN
    float* csrw = dinv + NN;                     // NE
    int*   ecnt = (int*)(csrw + NE);             // NN
    int*   offs = ecnt + NN;                     // NN+1
    int*   curs = offs + (NN + 1);               // NN
    int*   csrs = curs + NN;                     // NE
    (void)in_sizes; (void)n_in; (void)out_size; (void)ws_size;

    // graph normalization + deterministic CSR (one-time)
    k_zero   <<<196, 256, 0, stream>>>(ecnt, curs);
    k_count  <<<3125, 256, 0, stream>>>(ei, ecnt);
    k_dinv   <<<196, 256, 0, stream>>>(ecnt, dinv);
    k_scan   <<<1, 1024, 0, stream>>>(ecnt, offs);
    k_fill   <<<3125, 256, 0, stream>>>(ei, dinv, offs, curs, csrs, csrw);
    k_sortcsr<<<196, 256, 0, stream>>>(offs, csrs, csrw);

    // h = x0 = relu(X @ W1 + b1)
    k_lin1<<<391, 256, 0, stream>>>(x, w1, b1, h, x0);

    // 64 GCNII layers
    for (int l = 0; l < NLAYERS; ++l) {
        float beta = logf(1.0f / (float)(l + 1) + 1.0f);
        k_aggz      <<<6250, 256, 0, stream>>>(h, x0, dinv, offs, csrs, csrw, z);
        k_gemm_layer<<<391, 256, 0, stream>>>(z, cw + (size_t)l * HID * HID, h, beta);
    }

    // out = log_softmax(h @ W2 + b2)
    k_out<<<391, 128, 0, stream>>>(h, w2, b2, out);
}